// ComputeRawInstanceArea_60043642798478
// MI455X (gfx1250) — compile-verified
//
#include <hip/hip_runtime.h>
#include <hip/hip_bf16.h>

// Per-cell bin-overlap density gather (DREAMPlace-style raw instance area).
// Memory-bound: ~20 B/cell streaming (30 MB total -> ~1.3 us floor at
// 23.3 TB/s) + 9 gathers/cell into a 1 MB map that lives in the 192 MB L2.
// WMMA does not apply (3x3 bilinear form per cell); the MI455X-specific wins
// are NT cache hints on the stream (keep the map hot) and global_prefetch_b8.

typedef float f32x2 __attribute__((ext_vector_type(2)));

#define K_BINS   3
#define NUM_BINS 512
#define BIN_SZ   2.0f
#define INV_BIN  0.5f

__global__ __launch_bounds__(256) void raw_instance_area_kernel(
    const float* __restrict__ inst_pos,    // [NUM_CELLS, 2] interleaved x,y
    const float* __restrict__ inst_hsz,    // [NUM_CELLS, 2]
    const float* __restrict__ util,        // [512, 512] row-major (x outer, y inner)
    const int*   __restrict__ p_movable_lo,// device scalar
    float*       __restrict__ out,         // [n]
    int n)
{
    const int lo     = p_movable_lo[0];
    const int tid    = blockIdx.x * blockDim.x + threadIdx.x;
    const int stride = gridDim.x * blockDim.x;

    for (int t = tid; t < n; t += stride) {
        const int i = lo + t;

        // gfx1250 prefetch path (global_prefetch_b8): warm L2 with the next
        // grid-stride tile. Branch-free: clamp the prefetch index in-bounds
        // instead of guarding with exec-mask manipulation.
        {
            const int tp = min(t + stride, n - 1) + lo;
            __builtin_prefetch(inst_pos + 2 * (size_t)tp, 0, 1);
            __builtin_prefetch(inst_hsz + 2 * (size_t)tp, 0, 1);
        }

        // Non-temporal b64 loads: the 24 MB stream is read exactly once —
        // never let it evict the L2-resident utilization map.
        const f32x2 p = __builtin_nontemporal_load((const f32x2*)inst_pos + i);
        const f32x2 h = __builtin_nontemporal_load((const f32x2*)inst_hsz + i);

        const float xlo = p.x - h.x, xhi = p.x + h.x;
        const float ylo = p.y - h.y, yhi = p.y + h.y;

        // First bin containing the low edge (XL = YL = 0, bin = 2.0 exactly,
        // so *0.5f is bit-identical to /2.0f).
        const int bx0 = (int)floorf(xlo * INV_BIN);
        const int by0 = (int)floorf(ylo * INV_BIN);

        float ovx[K_BINS], ovy[K_BINS];
        int   bx[K_BINS],  by[K_BINS];

#pragma unroll
        for (int k = 0; k < K_BINS; ++k) {
            const int   b = bx0 + k;
            const float e = (float)b * BIN_SZ;
            float ov = fminf(xhi, e + BIN_SZ) - fmaxf(xlo, e);
            ov = fmaxf(ov, 0.0f);
            ovx[k] = ((unsigned)b < (unsigned)NUM_BINS) ? ov : 0.0f;
            bx[k]  = min(max(b, 0), NUM_BINS - 1);
        }
#pragma unroll
        for (int k = 0; k < K_BINS; ++k) {
            const int   b = by0 + k;
            const float e = (float)b * BIN_SZ;
            float ov = fminf(yhi, e + BIN_SZ) - fmaxf(ylo, e);
            ov = fmaxf(ov, 0.0f);
            ovy[k] = ((unsigned)b < (unsigned)NUM_BINS) ? ov : 0.0f;
            by[k]  = min(max(b, 0), NUM_BINS - 1);
        }

        // out[t] = sum_ij util[bx_i, by_j] * ovx_i * ovy_j
        // Gathers use default (regular-temporal) policy: map stays cached.
        float acc = 0.0f;
#pragma unroll
        for (int a = 0; a < K_BINS; ++a) {
            const float* __restrict__ row = util + bx[a] * NUM_BINS;
            float s =        row[by[0]] * ovy[0];
            s = fmaf(row[by[1]], ovy[1], s);
            s = fmaf(row[by[2]], ovy[2], s);
            acc = fmaf(ovx[a], s, acc);
        }

        // NT store: output is written once, never re-read by us.
        __builtin_nontemporal_store(acc, out + t);
    }
}

extern "C" void kernel_launch(void* const* d_in, const int* in_sizes, int n_in,
                              void* d_out, int out_size, void* d_ws, size_t ws_size,
                              hipStream_t stream) {
    const float* inst_pos = (const float*)d_in[0];   // [NUM_CELLS*2]
    const float* inst_hsz = (const float*)d_in[1];   // [NUM_CELLS*2]
    const float* util     = (const float*)d_in[2];   // [512*512]
    const int*   p_lo     = (const int*)  d_in[3];   // scalar movable_lo
    // d_in[4] = movable_hi (scalar); n = movable_hi - movable_lo == out_size.
    float*       out      = (float*)d_out;

    const int n = out_size;
    if (n <= 0) return;

    const int block = 256;                       // 8 wave32 waves per block
    int grid = (n + block - 1) / block;
    if (grid > 4096) grid = 4096;                // persistent-ish grid-stride:
                                                 // keeps prefetch distance useful
    raw_instance_area_kernel<<<grid, block, 0, stream>>>(
        inst_pos, inst_hsz, util, p_lo, out, n);
}